// GCLayer_23210003267969
// MI455X (gfx1250) — compile-verified
//
#include <hip/hip_runtime.h>
#include <hip/hip_bf16.h>
#include <math.h>

typedef float v2f __attribute__((ext_vector_type(2)));
typedef float v8f __attribute__((ext_vector_type(8)));

#if defined(__HIP_DEVICE_COMPILE__)
#if !__has_builtin(__builtin_amdgcn_wmma_f32_16x16x4_f32)
#error "wmma f32 16x16x4 builtin not available on this toolchain (device pass)"
#endif
#endif

#define NEG_SLOPE 0.01f
#define EPS_NORM 1e-12f
#define DCONST 128

// ---------------- zero helper ----------------
__global__ void zero_f32(float* __restrict__ p, size_t n) {
    size_t i = (size_t)blockIdx.x * blockDim.x + threadIdx.x;
    if (i < n) p[i] = 0.0f;
}

// ---------------- SPMM: ax[row] += val * x[col]  (one wave per edge) ----------------
__global__ void spmm_atomic(const int* __restrict__ erow, const int* __restrict__ ecol,
                            const float* __restrict__ eval, const float* __restrict__ x,
                            float* ax, int E) {
    int gtid = blockIdx.x * blockDim.x + threadIdx.x;
    int edge = gtid >> 5;
    int lane = gtid & 31;
    if (edge >= E) return;
    int r = erow[edge];
    int c = ecol[edge];
    float v = eval[edge];
    const float4* xp = (const float4*)(x + (size_t)c * DCONST);
    float4 xv = xp[lane];                       // coalesced 512B per wave
    float* dst = ax + (size_t)r * DCONST + lane * 4;
    atomicAdd(dst + 0, v * xv.x);
    atomicAdd(dst + 1, v * xv.y);
    atomicAdd(dst + 2, v * xv.z);
    atomicAdd(dst + 3, v * xv.w);
}

// ---------------- column sum of combine_new = [X[:M]; new_api] ----------------
__global__ void colsum_kernel(const float* __restrict__ former, const float* __restrict__ napi,
                              float* s, int N, int M) {
    int t = threadIdx.x;                        // 0..127 = column
    float acc = 0.0f;
    for (int row = blockIdx.x; row < N; row += gridDim.x) {
        const float* src = (row < M) ? (former + (size_t)row * DCONST)
                                     : (napi + (size_t)(row - M) * DCONST);
        acc += src[t];                          // coalesced across the 128 threads
    }
    atomicAdd(&s[t], acc);
}

// ---------------- key_sum -> v = Wq@key_sum, c0 = bq.key_sum (single block) ----------------
__global__ void attn_prep(const float* __restrict__ s, const float* __restrict__ Wk,
                          const float* __restrict__ bk, const float* __restrict__ Wq,
                          const float* __restrict__ bq, float* v, float* c0, float Nf) {
    __shared__ float ks[DCONST];
    __shared__ float red[DCONST];
    int t = threadIdx.x;
    float acc = 0.0f;
    for (int k = 0; k < DCONST; ++k) acc += s[k] * Wk[k * DCONST + t];
    acc += Nf * bk[t];
    ks[t] = acc;
    __syncthreads();
    float vv = 0.0f;
    for (int d = 0; d < DCONST; ++d) vv += Wq[t * DCONST + d] * ks[d];
    v[t] = vv;
    red[t] = bq[t] * ks[t];
    __syncthreads();
    for (int off = 64; off > 0; off >>= 1) {
        if (t < off) red[t] += red[t + off];
        __syncthreads();
    }
    if (t == 0) c0[0] = red[0];
}

// ---------------- fused: [ax+X | ax*X] @ [W0;W1] + (b0+b1), LeakyReLU, L2-normalize ----
// One 16-row tile per block iteration; 8 waves each own a 16-column slice.
// B fragments (256x16 slice of [W0;W1]) live in registers, loaded once per block.
// ax may alias out (rows are read before the same rows are written, per block).
__launch_bounds__(256)
__global__ void gc_fused(const float* ax, const float* __restrict__ x,
                         const float* __restrict__ W0, const float* __restrict__ b0,
                         const float* __restrict__ W1, const float* __restrict__ b1,
                         float* out, int nTiles) {
    __shared__ float Alds[16 * 260];            // 16 rows x 256 cols, stride 260 (no bank conflicts)
    __shared__ float sumsq[16];

    const int tid  = threadIdx.x;
    const int lane = tid & 31;
    const int w    = tid >> 5;                  // wave 0..7 -> column slice
    const int half = lane >> 4;                 // 0: K even pair, 1: K odd pair (per ISA A/B layout)
    const int l16  = lane & 15;
    const int col  = w * 16 + l16;              // global output column 0..127

    // ---- preload this wave's B fragments: rows 4*kk+2*half, +1 of [W0;W1], column `col`
    v2f bf[64];
#pragma unroll
    for (int kk = 0; kk < 64; ++kk) {
        const float* base;
        if (kk < 32) base = W0 + (4 * kk + 2 * half) * DCONST;
        else         base = W1 + (4 * (kk - 32) + 2 * half) * DCONST;
        bf[kk].x = base[col];
        bf[kk].y = base[DCONST + col];
    }
    const float bias = b0[col] + b1[col];

    for (int t = blockIdx.x; t < nTiles; t += gridDim.x) {
        __syncthreads();                        // previous tile fully consumed
        // ---- stage A tile: cols 0..127 = ax+x, cols 128..255 = ax*x
        for (int i = tid; i < 16 * DCONST; i += 256) {
            int r = i >> 7, c = i & 127;
            size_t g = ((size_t)t * 16 + r) * DCONST + c;
            float av = ax[g];
            float xv = x[g];
            Alds[r * 260 + c] = av + xv;
            Alds[r * 260 + 128 + c] = av * xv;
        }
        if (tid < 16) sumsq[tid] = 0.0f;
        // prefetch next tile while we compute
        {
            int tn = t + gridDim.x;
            if (tn < nTiles) {
                const float* pf = ax + ((size_t)tn * 16) * DCONST + tid * 8;
                __builtin_prefetch(pf, 0, 1);
                __builtin_prefetch(x + ((size_t)tn * 16) * DCONST + tid * 8, 0, 1);
            }
        }
        __syncthreads();

        // ---- K=256 accumulation: 64 chained V_WMMA_F32_16X16X4_F32
        v8f acc = {};
#if defined(__HIP_DEVICE_COMPILE__)
#pragma unroll
        for (int kk = 0; kk < 64; ++kk) {
            const float* ap = &Alds[l16 * 260 + 4 * kk + 2 * half];
            v2f af;
            af.x = ap[0];
            af.y = ap[1];
            acc = __builtin_amdgcn_wmma_f32_16x16x4_f32(
                false, af, false, bf[kk], (short)0, acc, false, false);
        }
#endif

        // ---- epilogue: bias, LeakyReLU, per-row sum of squares (16-lane shfl + LDS atomic)
        float vals[8];
#pragma unroll
        for (int r = 0; r < 8; ++r) {
            float vv = acc[r] + bias;
            vv = (vv > 0.0f) ? vv : NEG_SLOPE * vv;
            vals[r] = vv;
            float sq = vv * vv;
            sq += __shfl_xor(sq, 1);
            sq += __shfl_xor(sq, 2);
            sq += __shfl_xor(sq, 4);
            sq += __shfl_xor(sq, 8);            // lanes 0-15 and 16-31 reduce separately
            if (l16 == 0) atomicAdd(&sumsq[r + 8 * half], sq);
        }
        __syncthreads();

        // ---- normalize and store (row M = r + 8*half per ISA C/D layout)
#pragma unroll
        for (int r = 0; r < 8; ++r) {
            int grow = t * 16 + r + 8 * half;
            float nrm = sqrtf(sumsq[r + 8 * half]);
            float inv = 1.0f / fmaxf(nrm, EPS_NORM);
            out[(size_t)grow * DCONST + col] = vals[r] * inv;
        }
    }
}

// ---------------- att_out[i] = (X[i].v + c0) / sqrt(128)  (one wave per row) --------
__global__ void attn_out_kernel(const float* __restrict__ x, const float* __restrict__ v,
                                const float* __restrict__ c0, float* __restrict__ out, int N) {
    int gtid = blockIdx.x * blockDim.x + threadIdx.x;
    int row = gtid >> 5;
    int lane = gtid & 31;
    if (row >= N) return;
    const float4* xp = (const float4*)(x + (size_t)row * DCONST);
    const float4* vp = (const float4*)v;
    float4 xv = xp[lane];
    float4 vv = vp[lane];
    float d = xv.x * vv.x + xv.y * vv.y + xv.z * vv.z + xv.w * vv.w;
    d += __shfl_xor(d, 16);
    d += __shfl_xor(d, 8);
    d += __shfl_xor(d, 4);
    d += __shfl_xor(d, 2);
    d += __shfl_xor(d, 1);
    if (lane == 0) out[row] = (d + c0[0]) * 0.08838834764831845f;  // 1/sqrt(128)
}

extern "C" void kernel_launch(void* const* d_in, const int* in_sizes, int n_in,
                              void* d_out, int out_size, void* d_ws, size_t ws_size,
                              hipStream_t stream) {
    // input order per setup_inputs()
    const int*   edge_row = (const int*)  d_in[2];
    const int*   edge_col = (const int*)  d_in[3];
    const float* edge_val = (const float*)d_in[4];
    const float* former   = (const float*)d_in[5];
    const float* napi     = (const float*)d_in[6];
    const float* W0       = (const float*)d_in[7];
    const float* b0       = (const float*)d_in[8];
    const float* W1       = (const float*)d_in[9];
    const float* b1       = (const float*)d_in[10];
    const float* Wq       = (const float*)d_in[11];
    const float* bq       = (const float*)d_in[12];
    const float* Wk       = (const float*)d_in[13];
    const float* bk       = (const float*)d_in[14];

    const int E = in_sizes[2];
    const int N = in_sizes[5] / DCONST;
    const int M = N - in_sizes[6] / DCONST;

    float* part = (float*)d_out;                 // N*128, also serves as ax accumulator
    float* att  = part + (size_t)N * DCONST;     // N

    float* s  = (float*)d_ws;                    // 128
    float* v  = s + DCONST;                      // 128
    float* c0 = v + DCONST;                      // 1

    // 1) zero ax (in d_out) and s
    {
        size_t nz = (size_t)N * DCONST;
        zero_f32<<<(unsigned)((nz + 255) / 256), 256, 0, stream>>>(part, nz);
        zero_f32<<<1, 256, 0, stream>>>(s, (size_t)DCONST);
    }
    // 2) SPMM accumulate into part (= ax)
    {
        long long waves = E;
        unsigned blocks = (unsigned)((waves * 32 + 255) / 256);
        spmm_atomic<<<blocks, 256, 0, stream>>>(edge_row, edge_col, edge_val, former, part, E);
    }
    // 3) column sum of combine_new
    colsum_kernel<<<2048, 128, 0, stream>>>(former, napi, s, N, M);
    // 4) key_sum -> v, c0
    attn_prep<<<1, 128, 0, stream>>>(s, Wk, bk, Wq, bq, v, c0, (float)N);
    // 5) fused WMMA GC layer (in-place: reads ax rows, writes part rows per tile)
    {
        int nTiles = N / 16;                     // N = 200000 -> 12500 tiles
        gc_fused<<<1024, 256, 0, stream>>>(part, former, W0, b0, W1, b1, part, nTiles);
    }
    // 6) attention output
    {
        unsigned blocks = (unsigned)(((size_t)N * 32 + 255) / 256);
        attn_out_kernel<<<blocks, 256, 0, stream>>>(former, v, c0, att, N);
    }
}